// TransVLAD_43490838839368
// MI455X (gfx1250) — compile-verified
//
#include <hip/hip_runtime.h>
#include <hip/hip_bf16.h>

// TransVLAD forward for MI455X (gfx1250), all-fp32 via V_WMMA_F32_16X16X4_F32,
// with GLOBAL_LOAD_ASYNC_TO_LDS_B128 tile staging (ASYNCcnt-tracked DMA).
// Workspace layout (floats):
//   rnorm [N*P] | att [N*G*P] | wsumpart [N*G*K] | aggpart [N*G*K*D] | xe [N*E*P]
// Total ~137.2 MiB required in d_ws.

typedef __attribute__((ext_vector_type(2))) float v2f;
typedef __attribute__((ext_vector_type(8))) float v8f;
typedef int v4i_ __attribute__((vector_size(16)));  // matches builtin proto

#define N_ 32
#define C_ 512
#define P_ 1024
#define E_ 1024
#define G_ 8
#define D_ 128
#define K_ 64

#define OFF_RNORM 0
#define OFF_ATT   (N_ * P_)                     // 32768
#define OFF_WSUM  (OFF_ATT + N_ * G_ * P_)      // 294912
#define OFF_AGG   (OFF_WSUM + N_ * G_ * K_)     // 311296
#define OFF_XE    (OFF_AGG + N_ * G_ * K_ * D_) // 2408448

// ---- async memory->LDS staging (CDNA5 ASYNCcnt path), guarded --------------
#ifndef USE_ASYNC_LDS
#if defined(__has_builtin)
#if __has_builtin(__builtin_amdgcn_global_load_async_to_lds_b128) && \
    __has_builtin(__builtin_amdgcn_s_wait_asynccnt)
#define USE_ASYNC_LDS 1
#else
#define USE_ASYNC_LDS 0
#endif
#else
#define USE_ASYNC_LDS 0
#endif
#endif

#define AS3_ __attribute__((address_space(3)))

static __device__ __forceinline__ void copy16(const float* g, float* l) {
#if USE_ASYNC_LDS
  // 16B per lane, memory -> LDS DMA, tracked with ASYNCcnt.
  // Proto: (v4i* global_src, AS3 v4i* lds_dst, Imm offset, Imm cpol)
  __builtin_amdgcn_global_load_async_to_lds_b128((v4i_*)g, (AS3_ v4i_*)l, 0, 0);
#else
  *(float4*)l = *(const float4*)g;
#endif
}

static __device__ __forceinline__ void copy16_fence() {
#if USE_ASYNC_LDS
  __builtin_amdgcn_s_wait_asynccnt(0);  // s_wait_asynccnt 0
#endif
}

static __device__ __forceinline__ v8f wmma_f32(v2f a, v2f b, v8f c) {
  // D = A(16x4,f32) * B(4x16,f32) + C(16x16,f32)
  return __builtin_amdgcn_wmma_f32_16x16x4_f32(false, a, false, b, (short)0, c,
                                               false, false);
}

// ---------------- Kernel 1: per-pixel reciprocal L2 norm over channels -------
__global__ __launch_bounds__(256) void k_rnorm(const float* __restrict__ x,
                                               float* __restrict__ rnorm) {
  int id = blockIdx.x * 256 + threadIdx.x;  // n*P + p
  int n = id >> 10, p = id & (P_ - 1);
  const float* xp = x + (size_t)n * C_ * P_ + p;
  float s = 0.f;
#pragma unroll 4
  for (int c = 0; c < C_; ++c) {
    float v = xp[(size_t)c * P_];
    s += v * v;
  }
  rnorm[id] = 1.0f / fmaxf(sqrtf(s), 1e-12f);
}

// ---------------- Kernel 2: xe = (W1 @ x) * rnorm  (per image GEMM) ----------
// M=E=1024, N=P=1024, Kdim=C=512. Macro-tile 128x64; 8 waves each computing a
// 32x32 register tile (4 accumulators, 4 WMMA per 4 fragment loads).
__global__ __launch_bounds__(256) void k_gemm1(const float* __restrict__ x,
                                               const float* __restrict__ W1,
                                               const float* __restrict__ rnorm,
                                               float* __restrict__ xe) {
  __shared__ __align__(16) float As2[128][20];  // [m][kc], 80B rows (16B mult)
  __shared__ __align__(16) float Bs[16][64];    // [kc][p], 256B rows
  const int t = threadIdx.x;
  const int wave = t >> 5, lane = t & 31, half = lane >> 4, l16 = lane & 15;
  const int n = blockIdx.z;
  const int eBase = blockIdx.y * 128;
  const int pBase = blockIdx.x * 64;
  const int mW = (wave & 3) * 32;   // 4 m-positions
  const int nW = (wave >> 2) * 32;  // 2 n-positions
  const float* xn = x + (size_t)n * C_ * P_;
  v8f acc[2][2] = {};
  for (int cBase = 0; cBase < C_; cBase += 16) {
    // stage A: W1[eBase..+128, cBase..+16] -> As2[m][kk] (16B per lane)
#pragma unroll
    for (int i = 0; i < 2; ++i) {
      int idx = t + i * 256;  // 0..511
      int kk4 = idx & 3, m = idx >> 2;
      copy16(&W1[(size_t)(eBase + m) * C_ + cBase + kk4 * 4], &As2[m][kk4 * 4]);
    }
    // stage B: x[n, cBase..+16, pBase..+64] -> Bs[kk][pp]
    {
      int pp4 = t & 15, kk = t >> 4;
      copy16(&xn[(size_t)(cBase + kk) * P_ + pBase + pp4 * 4], &Bs[kk][pp4 * 4]);
    }
    copy16_fence();
    __syncthreads();
#pragma unroll
    for (int kk = 0; kk < 16; kk += 4) {
      v2f a0, a1, b0, b1;
      a0.x = As2[mW + l16][kk + 2 * half];
      a0.y = As2[mW + l16][kk + 2 * half + 1];
      a1.x = As2[mW + 16 + l16][kk + 2 * half];
      a1.y = As2[mW + 16 + l16][kk + 2 * half + 1];
      b0.x = Bs[kk + 2 * half][nW + l16];
      b0.y = Bs[kk + 2 * half + 1][nW + l16];
      b1.x = Bs[kk + 2 * half][nW + 16 + l16];
      b1.y = Bs[kk + 2 * half + 1][nW + 16 + l16];
      acc[0][0] = wmma_f32(a0, b0, acc[0][0]);
      acc[0][1] = wmma_f32(a0, b1, acc[0][1]);
      acc[1][0] = wmma_f32(a1, b0, acc[1][0]);
      acc[1][1] = wmma_f32(a1, b1, acc[1][1]);
    }
    __syncthreads();  // all readers done before next stage overwrites
  }
#pragma unroll
  for (int ms = 0; ms < 2; ++ms)
#pragma unroll
    for (int ns = 0; ns < 2; ++ns) {
      const float rn = rnorm[n * P_ + pBase + nW + ns * 16 + l16];
#pragma unroll
      for (int v = 0; v < 8; ++v) {
        int e = eBase + mW + ms * 16 + v + 8 * half;
        xe[((size_t)n * E_ + e) * P_ + pBase + nW + ns * 16 + l16] =
            acc[ms][ns][v] * rn;
      }
    }
}

// ---------------- Kernel 3: att = sigmoid(W2 @ xe), G=8 ---------------------
__global__ __launch_bounds__(256) void k_att(const float* __restrict__ xe,
                                             const float* __restrict__ W2,
                                             float* __restrict__ att) {
  __shared__ float s_W2[G_ * E_];
  const int t = threadIdx.x;
#pragma unroll
  for (int i = 0; i < (G_ * E_) / 256; ++i) s_W2[t + i * 256] = W2[t + i * 256];
  __syncthreads();
  int id = blockIdx.x * 256 + t;  // n*P + p
  int n = id >> 10, p = id & (P_ - 1);
  const float* xep = xe + (size_t)n * E_ * P_ + p;
  float z[G_];
#pragma unroll
  for (int g = 0; g < G_; ++g) z[g] = 0.f;
  for (int e = 0; e < E_; ++e) {
    float v = xep[(size_t)e * P_];
#pragma unroll
    for (int g = 0; g < G_; ++g) z[g] += s_W2[g * E_ + e] * v;
  }
#pragma unroll
  for (int g = 0; g < G_; ++g)
    att[((size_t)n * G_ + g) * P_ + p] = 1.0f / (1.0f + expf(-z[g]));
}

// ---------------- Kernel 4: soft-assign + weighted aggregation --------------
// One workgroup per (n,g). Per 32-pixel tile:
//   logits[64x32] = Wc[64x128] @ xg[128x32]   (WMMA)
//   softmax over K, scale by att -> w
//   agg[64x128] += w[64x32] @ xg^T[32x128]    (WMMA, accumulated in regs)
__global__ __launch_bounds__(256) void k_vlad(const float* __restrict__ xe,
                                              const float* __restrict__ Wc,
                                              const float* __restrict__ att,
                                              float* __restrict__ aggpart,
                                              float* __restrict__ wsumpart) {
  __shared__ float WcA[D_][65];                // [d][k], padded
  __shared__ __align__(16) float Xg[D_][36];   // [d][p], 144B rows (16B mult)
  __shared__ float Lg[K_][33];                 // [k][p] logits then weights
  __shared__ float wsumAcc[K_];
  const int t = threadIdx.x;
  const int wave = t >> 5, lane = t & 31, half = lane >> 4, l16 = lane & 15;
  const int g = blockIdx.x, n = blockIdx.y;
#pragma unroll
  for (int i = 0; i < (K_ * D_) / 256; ++i) {  // Wc -> LDS transposed
    int idx = t + i * 256;
    int d = idx & 127, k = idx >> 7;
    WcA[d][k] = Wc[k * D_ + d];
  }
  if (t < K_) wsumAcc[t] = 0.f;
  const int kL = (wave & 3) * 16, pL = (wave >> 2) * 16;  // logits subtile
  const int kA = (wave >> 1) * 16, dA = (wave & 1) * 64;  // agg strip
  v8f aggAcc[4] = {};
  const float* xg_base = xe + ((size_t)n * E_ + g * D_) * P_;
  const float* att_base = att + ((size_t)n * G_ + g) * P_;

  for (int ptile = 0; ptile < P_ / 32; ++ptile) {
    const int pBase = ptile * 32;
    __syncthreads();  // previous tile's readers done
#pragma unroll
    for (int i = 0; i < 4; ++i) {  // stage xg tile (16B per lane)
      int idx = t + i * 256;       // 0..1023
      int pp4 = idx & 7, d = idx >> 3;
      copy16(&xg_base[(size_t)d * P_ + pBase + pp4 * 4], &Xg[d][pp4 * 4]);
    }
    copy16_fence();
    __syncthreads();
    // logits GEMM
    v8f acc = {};
#pragma unroll 4
    for (int dd = 0; dd < D_; dd += 4) {
      v2f a, b;
      a.x = WcA[dd + 2 * half][kL + l16];
      a.y = WcA[dd + 2 * half + 1][kL + l16];
      b.x = Xg[dd + 2 * half][pL + l16];
      b.y = Xg[dd + 2 * half + 1][pL + l16];
      acc = wmma_f32(a, b, acc);
    }
#pragma unroll
    for (int v = 0; v < 8; ++v) Lg[kL + v + 8 * half][pL + l16] = acc[v];
    __syncthreads();
    // softmax over K per pixel column, fold in attention gate
    if (t < 32) {
      const int p = t;
      float m = -1e30f;
      for (int k = 0; k < K_; ++k) m = fmaxf(m, Lg[k][p]);
      float s = 0.f;
      for (int k = 0; k < K_; ++k) {
        float e = expf(Lg[k][p] - m);
        Lg[k][p] = e;
        s += e;
      }
      const float sc = att_base[pBase + p] / s;
      for (int k = 0; k < K_; ++k) Lg[k][p] *= sc;
    }
    __syncthreads();
    // wsum accumulation (deterministic, one thread per k)
    if (t < K_) {
      float s = 0.f;
#pragma unroll
      for (int p = 0; p < 32; ++p) s += Lg[t][p];
      wsumAcc[t] += s;
    }
    // aggregate GEMM: agg += w @ xg^T, Kdim = 32 pixels
#pragma unroll
    for (int pp = 0; pp < 32; pp += 4) {
      v2f a;
      a.x = Lg[kA + l16][pp + 2 * half];
      a.y = Lg[kA + l16][pp + 2 * half + 1];
#pragma unroll
      for (int j = 0; j < 4; ++j) {
        v2f b;
        b.x = Xg[dA + j * 16 + l16][pp + 2 * half];
        b.y = Xg[dA + j * 16 + l16][pp + 2 * half + 1];
        aggAcc[j] = wmma_f32(a, b, aggAcc[j]);
      }
    }
  }
  __syncthreads();
  float* aggp = aggpart + (size_t)(n * G_ + g) * K_ * D_;
#pragma unroll
  for (int j = 0; j < 4; ++j)
#pragma unroll
    for (int v = 0; v < 8; ++v) {
      int k = kA + v + 8 * half;
      int d = dA + j * 16 + l16;
      aggp[k * D_ + d] = aggAcc[j][v];
    }
  if (t < K_) wsumpart[(size_t)(n * G_ + g) * K_ + t] = wsumAcc[t];
}

// ---------------- Kernel 5: reduce groups, subtract centroid term -----------
__global__ __launch_bounds__(256) void k_final(const float* __restrict__ aggpart,
                                               const float* __restrict__ wsumpart,
                                               const float* __restrict__ centroids,
                                               float* __restrict__ out) {
  int id = blockIdx.x * 256 + threadIdx.x;  // n*(K*D) + k*D + d
  int n = id >> 13;
  int r = id & (K_ * D_ - 1);
  int k = r >> 7;
  float s = 0.f, wsum = 0.f;
#pragma unroll
  for (int g = 0; g < G_; ++g) {
    s += aggpart[(size_t)(n * G_ + g) * K_ * D_ + r];
    wsum += wsumpart[(size_t)(n * G_ + g) * K_ + k];
  }
  out[id] = s - wsum * centroids[r];
}

extern "C" void kernel_launch(void* const* d_in, const int* in_sizes, int n_in,
                              void* d_out, int out_size, void* d_ws, size_t ws_size,
                              hipStream_t stream) {
  (void)in_sizes; (void)n_in; (void)out_size; (void)ws_size;
  const float* x = (const float*)d_in[0];
  const float* W1 = (const float*)d_in[1];
  const float* W2 = (const float*)d_in[2];
  const float* Wc = (const float*)d_in[3];
  const float* centroids = (const float*)d_in[4];
  float* ws = (float*)d_ws;
  float* rnorm = ws + OFF_RNORM;
  float* attw  = ws + OFF_ATT;
  float* wsump = ws + OFF_WSUM;
  float* aggp  = ws + OFF_AGG;
  float* xe    = ws + OFF_XE;
  float* out = (float*)d_out;

  k_rnorm<<<(N_ * P_) / 256, 256, 0, stream>>>(x, rnorm);
  k_gemm1<<<dim3(P_ / 64, E_ / 128, N_), 256, 0, stream>>>(x, W1, rnorm, xe);
  k_att<<<(N_ * P_) / 256, 256, 0, stream>>>(xe, W2, attw);
  k_vlad<<<dim3(G_, N_), 256, 0, stream>>>(xe, Wc, attw, aggp, wsump);
  k_final<<<(N_ * K_ * D_) / 256, 256, 0, stream>>>(aggp, wsump, centroids, out);
}